// TransD_48361331753003
// MI455X (gfx1250) — compile-verified
//
#include <hip/hip_runtime.h>

// TransD margin loss for MI455X (gfx1250, wave32).
// Memory-bound gather kernel: ~200MB of gathered embedding rows, ~108 MFLOP.
// Strategy: 1 wave per batch index, fully-coalesced b128 row gathers,
// cross-lane reductions via V_WMMA_F32_16X16X4_F32 (ones-matrix trick),
// global_prefetch_b8 to hide the neg-triple gather latency.

typedef float v2f __attribute__((ext_vector_type(2)));
typedef float v8f __attribute__((ext_vector_type(8)));

#define DIM 256
#define MARGIN_F 1.0f
#define EPS_F 1e-6f

// 32-lane f32 sum using V_WMMA_F32_16X16X4_F32 with B == ones.
// With B all-ones, D[m][n] = sum_k A[m][k] for every n, independent of the
// exact lane<->(m,k) mapping. Putting x in A's VGPR0 and 0 in VGPR1 means each
// lane's value contributes exactly once. Lanes 0-15 hold rows 0-7 of a column
// of D, lanes 16-31 hold rows 8-15, so per-lane sum of the 8 accumulator VGPRs
// + one shfl_xor(16) yields the full 32-lane sum in every lane.
__device__ __forceinline__ float wave_reduce_add(float x) {
  v2f a; a[0] = x;    a[1] = 0.0f;
  v2f b; b[0] = 1.0f; b[1] = 1.0f;
  v8f c = {};
  v8f d = __builtin_amdgcn_wmma_f32_16x16x4_f32(
      /*neg_a=*/false, a, /*neg_b=*/false, b,
      /*c_mod=*/(short)0, c, /*reuse_a=*/false, /*reuse_b=*/false);
  float s = ((d[0] + d[1]) + (d[2] + d[3])) + ((d[4] + d[5]) + (d[6] + d[7]));
  s += __shfl_xor(s, 16, 32);
  return s;
}

__device__ __forceinline__ float dot4(float4 a, float4 b) {
  float r = a.x * b.x;
  r = fmaf(a.y, b.y, r);
  r = fmaf(a.z, b.z, r);
  r = fmaf(a.w, b.w, r);
  return r;
}

// One wave computes the TransD score of one (h, r, t) triple.
// Lane l owns elements [4l,4l+4) and [512B + 4l, ...): each b128 load covers a
// contiguous 512B half-row across the wave -> perfectly coalesced gathers.
__device__ __forceinline__ float triple_score(
    const float* __restrict__ eemb, const float* __restrict__ emap,
    const float* __restrict__ remb, const float* __restrict__ rmap,
    int h, int r, int t, int lane) {
  const float4* hrow  = (const float4*)(eemb + (size_t)h * DIM);
  const float4* hprow = (const float4*)(emap + (size_t)h * DIM);
  const float4* trow  = (const float4*)(eemb + (size_t)t * DIM);
  const float4* tprow = (const float4*)(emap + (size_t)t * DIM);
  const float4* rrow  = (const float4*)(remb + (size_t)r * DIM);
  const float4* rprow = (const float4*)(rmap + (size_t)r * DIM);

  const int i0 = lane, i1 = lane + 32;   // DIM/4 = 64 float4 per row

  float4 hv0 = hrow[i0],  hv1 = hrow[i1];
  float4 hp0 = hprow[i0], hp1 = hprow[i1];
  float4 tv0 = trow[i0],  tv1 = trow[i1];
  float4 tp0 = tprow[i0], tp1 = tprow[i1];
  float4 rv0 = rrow[i0],  rv1 = rrow[i1];
  float4 rp0 = rprow[i0], rp1 = rprow[i1];

  // dot(hp,h) and dot(tp,t): per-lane partial, then wave-wide WMMA reduce.
  float dh = wave_reduce_add(dot4(hp0, hv0) + dot4(hp1, hv1));
  float dt = wave_reduce_add(dot4(tp0, tv0) + dot4(tp1, tv1));

  // diff = (rp*dh + h) + r - (rp*dt + t) + eps, accumulated as sum of squares.
  float acc = 0.0f;
#define ACC_ELEM(RP, HV, TV, RV)                          \
  {                                                       \
    float hpj = fmaf((RP), dh, (HV));                     \
    float tpj = fmaf((RP), dt, (TV));                     \
    float df  = ((hpj + (RV)) - tpj) + EPS_F;             \
    acc = fmaf(df, df, acc);                              \
  }
  ACC_ELEM(rp0.x, hv0.x, tv0.x, rv0.x)
  ACC_ELEM(rp0.y, hv0.y, tv0.y, rv0.y)
  ACC_ELEM(rp0.z, hv0.z, tv0.z, rv0.z)
  ACC_ELEM(rp0.w, hv0.w, tv0.w, rv0.w)
  ACC_ELEM(rp1.x, hv1.x, tv1.x, rv1.x)
  ACC_ELEM(rp1.y, hv1.y, tv1.y, rv1.y)
  ACC_ELEM(rp1.z, hv1.z, tv1.z, rv1.z)
  ACC_ELEM(rp1.w, hv1.w, tv1.w, rv1.w)
#undef ACC_ELEM

  float s2 = wave_reduce_add(acc);
  return sqrtf(s2);
}

// Grid: one wave per batch index; 256-thread blocks = 8 waves.
__global__ void transd_pair_kernel(
    const int* __restrict__ pos_x, const int* __restrict__ neg_x,
    const float* __restrict__ eemb, const float* __restrict__ emap,
    const float* __restrict__ remb, const float* __restrict__ rmap,
    float* __restrict__ partial, int batch) {
  const int wave = (int)((blockIdx.x * blockDim.x + threadIdx.x) >> 5);
  const int lane = (int)(threadIdx.x & 31);
  if (wave >= batch) return;  // wave-uniform: EXEC stays all-ones for WMMA

  const int ph = pos_x[wave * 3 + 0];
  const int pr = pos_x[wave * 3 + 1];
  const int pt = pos_x[wave * 3 + 2];
  const int nh = neg_x[wave * 3 + 0];
  const int nr = neg_x[wave * 3 + 1];
  const int nt = neg_x[wave * 3 + 2];

  // Prefetch the negative-triple entity rows (4 x 1KB) while the positive
  // triple is fetched and scored. Lane stride 32B covers each full row.
  {
    const float* p0 = eemb + (size_t)nh * DIM + lane * 8;
    const float* p1 = emap + (size_t)nh * DIM + lane * 8;
    const float* p2 = eemb + (size_t)nt * DIM + lane * 8;
    const float* p3 = emap + (size_t)nt * DIM + lane * 8;
    __builtin_prefetch(p0, 0, 0);
    __builtin_prefetch(p1, 0, 0);
    __builtin_prefetch(p2, 0, 0);
    __builtin_prefetch(p3, 0, 0);
  }

  float ps = triple_score(eemb, emap, remb, rmap, ph, pr, pt, lane);
  float ns = triple_score(eemb, emap, remb, rmap, nh, nr, nt, lane);

  float v = ps - ns + MARGIN_F;
  v = v > 0.0f ? v : 0.0f;
  if (lane == 0) partial[wave] = v;
}

// Deterministic final reduction: fixed summation order every call.
__global__ void transd_reduce_kernel(const float* __restrict__ partial,
                                     float* __restrict__ out, int n) {
  const int tid = (int)threadIdx.x;
  float p = 0.0f;
  for (int j = tid; j < n; j += 256) p += partial[j];

  float wsum = wave_reduce_add(p);  // all 256 lanes active -> EXEC all-ones

  __shared__ float s[8];
  const int lane = tid & 31, w = tid >> 5;
  if (lane == 0) s[w] = wsum;
  __syncthreads();
  if (tid == 0) {
    float t = 0.0f;
    for (int k = 0; k < 8; ++k) t += s[k];
    out[0] = t / (float)n;
  }
}

extern "C" void kernel_launch(void* const* d_in, const int* in_sizes, int n_in,
                              void* d_out, int out_size, void* d_ws, size_t ws_size,
                              hipStream_t stream) {
  const int*   pos_x = (const int*)d_in[0];
  const int*   neg_x = (const int*)d_in[1];
  const float* eemb  = (const float*)d_in[2];
  const float* emap  = (const float*)d_in[3];
  const float* remb  = (const float*)d_in[4];
  const float* rmap  = (const float*)d_in[5];

  const int batch = in_sizes[0] / 3;          // 16384
  float* partial = (float*)d_ws;              // batch floats of scratch

  const int blocks = (batch + 7) / 8;         // 8 waves (256 threads) per block
  transd_pair_kernel<<<blocks, 256, 0, stream>>>(pos_x, neg_x, eemb, emap,
                                                 remb, rmap, partial, batch);
  transd_reduce_kernel<<<1, 256, 0, stream>>>(partial, (float*)d_out, batch);
}